// SpectralConv1dOptimized_51376398795179
// MI455X (gfx1250) — compile-verified
//
#include <hip/hip_runtime.h>
#include <math.h>

#define PI_D 3.14159265358979323846

typedef float v2f __attribute__((ext_vector_type(2)));
typedef float v8f __attribute__((ext_vector_type(8)));

// Problem constants (from reference)
constexpr int B_     = 8;
constexpr int CIN_   = 64;
constexpr int COUT_  = 64;
constexpr int N_     = 4096;
constexpr int MODES_ = 64;
constexpr int MCH_   = 20;             // Chebyshev points per segment (n = 19)
constexpr int LM_    = 100 * MCH_;     // 2000 quadrature nodes
constexpr int BC_    = B_ * CIN_;      // 512 rows
constexpr int KC_    = 128;            // stacked re/im (or cos/sin) width

// Clenshaw-Curtis weight for node k on [-1,1], m=20 (n=19, odd -> b always 2)
__device__ __forceinline__ double cc_weight(int k) {
  double v = 1.0;
#pragma unroll
  for (int j = 1; j <= 9; ++j)
    v -= 2.0 / (4.0 * j * j - 1.0) * cos(2.0 * j * k * PI_D / 19.0);
  double w = 2.0 * v / 19.0;
  if (k == 0 || k == 19) w *= 0.5;
  return w;
}

// ---------------- K1: interpolate x onto time quadrature nodes --------------
__global__ __launch_bounds__(256)
void k_interp_xs(const float* __restrict__ x, float* __restrict__ xs) {
  int q = blockIdx.y * 256 + threadIdx.x;
  if (q >= LM_) return;
  int row = blockIdx.x;                 // 0..511 (b*CIN + cin)
  int l = q / MCH_, k = q % MCH_;
  double node = cos(PI_D * k / 19.0);
  double tq   = (l + 0.5) / 100.0 + 0.005 * node;   // cents + h*node, h=0.5/L
  double pos  = tq * 4095.0;                         // t = linspace(0,1,4096)
  int j = (int)floor(pos);
  if (j < 0) j = 0;
  if (j > 4094) j = 4094;
  float fr = (float)(pos - (double)j);
  const float* xr = x + (size_t)row * N_;
  xs[(size_t)row * LM_ + q] = xr[j] + (xr[j + 1] - xr[j]) * fr;
}

// ----- K2: CFT basis, TRANSPOSED Et (128 x 2000): [wts*cos | -wts*sin] -----
__global__ __launch_bounds__(256)
void k_build_Et(float* __restrict__ Et) {
  int idx = blockIdx.x * 256 + threadIdx.x;
  if (idx >= KC_ * LM_) return;
  int c = idx / LM_;           // output column of the GEMM (0..127)
  int q = idx - c * LM_;       // quadrature node (0..1999)
  int l = q / MCH_, k = q % MCH_;
  double node = cos(PI_D * k / 19.0);
  double tq   = (l + 0.5) / 100.0 + 0.005 * node;
  double wt   = 0.005 * cc_weight(k);     // h * cw
  int freq = c & 63;                      // f = 0..63 (fft order head)
  double ph = 2.0 * PI_D * (double)freq * tq;
  double s, cc;
  sincos(ph, &s, &cc);
  // X = xs @ E : exp(-i ph) * wt  ->  re = wt*cos, im = -wt*sin
  Et[idx] = (float)((c < 64) ? wt * cc : -wt * s);
}

// ----- K4: complex spectral channel mixing  out[b,o,m] = sum_i X*W ----------
__global__ __launch_bounds__(256)
void k_mix(const float* __restrict__ Xcat, const float* __restrict__ wr,
           const float* __restrict__ wi, float* __restrict__ Yr,
           float* __restrict__ Yi) {
  int idx = blockIdx.x * 256 + threadIdx.x;      // 512*64
  if (idx >= BC_ * MODES_) return;
  int m  = idx & 63;
  int bo = idx >> 6;
  int b  = bo >> 6;
  int o  = bo & 63;
  float sr = 0.f, si = 0.f;
  for (int i = 0; i < CIN_; ++i) {
    const float* xrow = Xcat + ((size_t)(b * CIN_ + i)) * KC_;
    float xr = xrow[m];
    float xi = xrow[64 + m];
    size_t wofs = ((size_t)i * COUT_ + o) * MODES_ + m;
    float wrv = wr[wofs];
    float wiv = wi[wofs];
    sr += xr * wrv - xi * wiv;
    si += xr * wiv + xi * wrv;
  }
  Yr[idx] = sr;
  Yi[idx] = si;
}

// ----- K5: ICFT A-matrix (512 x 128): [wfs*ci_re | -wfs*ci_im] --------------
// Only freq-quadrature nodes gq in [980,1040) (segments 49..51) can be
// nonzero: the spectrum is zero outside f in (-1, 64). qLoc 60..63 = zero pad.
__global__ __launch_bounds__(256)
void k_build_A(const float* __restrict__ Yr, const float* __restrict__ Yi,
               float* __restrict__ Acat) {
  int idx = blockIdx.x * 256 + threadIdx.x;      // 512*128
  if (idx >= BC_ * KC_) return;
  int c    = idx & 127;
  int bo   = idx >> 7;
  int qLoc = c & 63;
  float val = 0.0f;
  if (qLoc < 60) {
    int gq = 980 + qLoc;
    int l = gq / MCH_, k = gq % MCH_;
    double node = cos(PI_D * k / 19.0);
    double hf   = 4095.0 / 200.0;                       // (fb-fa)/(2L)
    double fqv  = -2048.0 + (l + 0.5) * (4095.0 / 100.0) + hf * node;
    double wf   = hf * cc_weight(k);
    int j = (int)floor(fqv);
    float fr = (float)(fqv - (double)j);
    const float* Y = (c < 64) ? Yr : Yi;
    float y0 = (j >= 0 && j < 64) ? Y[(size_t)bo * 64 + j] : 0.0f;
    float y1 = (j + 1 >= 0 && j + 1 < 64) ? Y[(size_t)bo * 64 + j + 1] : 0.0f;
    float ci = y0 + (y1 - y0) * fr;                     // linear interp
    val = (float)wf * ci;
    if (c >= 64) val = -val;  // Re(ci * e^{+i ph}) = cr*cos - cii*sin
  }
  Acat[idx] = val;
}

// -- K6: ICFT basis, TRANSPOSED Bt (4096 x 128): [cos(2pi fq t) | sin(...)] --
__global__ __launch_bounds__(256)
void k_build_Bt(float* __restrict__ Bt) {
  int idx = blockIdx.x * 256 + threadIdx.x;      // 4096*128
  if (idx >= N_ * KC_) return;
  int t = idx >> 7;            // time sample 0..4095
  int r = idx & 127;           // GEMM K-row 0..127
  int qLoc = r & 63;
  float val = 0.0f;
  if (qLoc < 60) {
    int gq = 980 + qLoc;
    int l = gq / MCH_, k = gq % MCH_;
    double node = cos(PI_D * k / 19.0);
    double fqv  = -2048.0 + (l + 0.5) * (4095.0 / 100.0) + (4095.0 / 200.0) * node;
    double tv   = (double)t / 4095.0;
    double ph = 2.0 * PI_D * fqv * tv;
    double s, cc;
    sincos(ph, &s, &cc);
    val = (float)((r < 64) ? cc : s);
  }
  Bt[idx] = val;
}

// - WMMA f32 GEMM (B transposed): C(MxN) = A(MxK) @ BT(NxK)^T, K%4==0 -------
// One wave per 16x16 C tile; both fragments contiguous in K -> b64 loads.
// A frag (16x4):  lane L: M = L%16, VGPR v -> K = kbase + 2*(L/16) + v
// B frag (4x16):  lane L: N = L%16, VGPR v -> K = kbase + 2*(L/16) + v
// C/D (16x16):    lane L: N = L%16, VGPR v -> M = 8*(L/16) + v
__global__ __launch_bounds__(256)
void k_wmma_gemm_tn(const float* __restrict__ A, const float* __restrict__ BT,
                    float* __restrict__ C, int K, int N, int nTilesN) {
  int wave = (blockIdx.x << 3) + (threadIdx.x >> 5);
  int lane = threadIdx.x & 31;
  int half = lane >> 4;
  int ln   = lane & 15;
  int tm = wave / nTilesN;
  int tn = wave - tm * nTilesN;

  // fold the lane-half K offset (2 floats = 1 v2f) into the base pointers
  const v2f* ap =
      (const v2f*)(A + (size_t)(tm * 16 + ln) * K + (half << 1));
  const v2f* bp =
      (const v2f*)(BT + (size_t)(tn * 16 + ln) * K + (half << 1));

  v8f acc = {0.f, 0.f, 0.f, 0.f, 0.f, 0.f, 0.f, 0.f};
  for (int k0 = 0; k0 < K; k0 += 4) {
    v2f a = ap[k0 >> 1];
    v2f b = bp[k0 >> 1];
    // D = A x B + C, f32 16x16x4
    acc = __builtin_amdgcn_wmma_f32_16x16x4_f32(
        /*neg_a=*/false, a, /*neg_b=*/false, b,
        /*c_mod=*/(short)0, acc, /*reuse_a=*/false, /*reuse_b=*/false);
  }

  float* crow = C + (size_t)(tm * 16 + (half << 3)) * N + (tn * 16 + ln);
#pragma unroll
  for (int v = 0; v < 8; ++v) crow[(size_t)v * N] = acc[v];
}

extern "C" void kernel_launch(void* const* d_in, const int* in_sizes, int n_in,
                              void* d_out, int out_size, void* d_ws,
                              size_t ws_size, hipStream_t stream) {
  (void)in_sizes; (void)n_in; (void)out_size; (void)ws_size;
  const float* x  = (const float*)d_in[0];   // (8, 64, 4096)
  const float* wr = (const float*)d_in[1];   // (64, 64, 64)
  const float* wi = (const float*)d_in[2];   // (64, 64, 64)
  float* out = (float*)d_out;                // (8, 64, 4096)

  // workspace layout (floats), total ~8.0 MB
  float* xs   = (float*)d_ws;                     // 512*2000
  float* Et   = xs + (size_t)BC_ * LM_;           // 128*2000  (transposed E)
  float* Xcat = Et + (size_t)KC_ * LM_;           // 512*128
  float* Yr   = Xcat + (size_t)BC_ * KC_;         // 512*64
  float* Yi   = Yr + (size_t)BC_ * MODES_;        // 512*64
  float* Acat = Yi + (size_t)BC_ * MODES_;        // 512*128
  float* Bt   = Acat + (size_t)BC_ * KC_;         // 4096*128  (transposed B)

  // 1) interp x -> xs
  k_interp_xs<<<dim3(BC_, (LM_ + 255) / 256), 256, 0, stream>>>(x, xs);
  // 2) CFT basis (transposed)
  k_build_Et<<<(KC_ * LM_ + 255) / 256, 256, 0, stream>>>(Et);
  // 3) Xcat = xs @ Et^T   (512 x 2000) x (2000 x 128); 256 tiles, 8/block
  k_wmma_gemm_tn<<<(BC_ / 16) * (KC_ / 16) / 8, 256, 0, stream>>>(
      xs, Et, Xcat, LM_, KC_, KC_ / 16);
  // 4) spectral channel mixing
  k_mix<<<(BC_ * MODES_ + 255) / 256, 256, 0, stream>>>(Xcat, wr, wi, Yr, Yi);
  // 5) ICFT A-matrix (sparse freq-quadrature interpolation + weights)
  k_build_A<<<(BC_ * KC_ + 255) / 256, 256, 0, stream>>>(Yr, Yi, Acat);
  // 6) ICFT cos/sin basis (transposed)
  k_build_Bt<<<(N_ * KC_ + 255) / 256, 256, 0, stream>>>(Bt);
  // 7) out = Acat @ Bt^T  (512 x 128) x (128 x 4096); 8192 tiles, 8/block
  k_wmma_gemm_tn<<<(BC_ / 16) * (N_ / 16) / 8, 256, 0, stream>>>(
      Acat, Bt, out, KC_, N_, N_ / 16);
}